// GCN_3470333575495
// MI455X (gfx1250) — compile-verified
//
#include <hip/hip_runtime.h>

// CDNA5 wave32 WMMA types
typedef __attribute__((ext_vector_type(2))) float v2f;
typedef __attribute__((ext_vector_type(8))) float v8f;

// ---------------------------------------------------------------------------
// Pre-transpose + zero-pad weights: Wt[n][k] = (n < Nreal) ? W[k][n] : 0
// Makes every B fragment a contiguous (K-adjacent) 8-byte global load.
// ---------------------------------------------------------------------------
__global__ void gcn_transpose_pad(const float* __restrict__ W, float* __restrict__ Wt,
                                  int K, int Nreal, int Npad)
{
    int idx = blockIdx.x * blockDim.x + threadIdx.x;
    if (idx >= Npad * K) return;
    int n = idx / K;
    int k = idx - n * K;
    Wt[idx] = (n < Nreal) ? W[(size_t)k * Nreal + n] : 0.f;
}

// ---------------------------------------------------------------------------
// GEMM + norm-scale, f32 WMMA 16x16x4.
// One wave computes a 16-row x (NT*16)-col output stripe.
// A frag (32-bit A 16x4): lane<16 holds A[m][0..1], lane>=16 holds A[m][2..3]
// B frag (4x16):          lane<16 holds B[0..1][n], lane>=16 holds B[2..3][n]
//   -> from Wt[n][k] these are 2 contiguous floats: one global_load_b64.
// C/D: vgpr r -> row r (lanes 0-15) / row r+8 (lanes 16-31), col = lane&15
// ---------------------------------------------------------------------------
template <int NT>
__global__ __launch_bounds__(128) void gcn_gemm_norm_wmma(
    const float* __restrict__ A,      // [M, K]
    const float* __restrict__ Wt,     // [NT*16, K] transposed, zero-padded
    const float* __restrict__ normv,  // [M]
    float* __restrict__ out,          // [M, NT*16]
    int M, int K)
{
    constexpr int Npad = NT * 16;
    __shared__ float sA[4][16 * 64];  // per-wave 16x64 A chunk (16 KB total)

    const int lane = threadIdx.x & 31;
    const int wave = threadIdx.x >> 5;
    const int gw = blockIdx.x * 4 + wave;
    const int mTiles = (M + 15) >> 4;
    if (gw >= mTiles) return;  // wave-uniform: EXEC stays all-1s for WMMA
    const int m0 = gw << 4;

    v8f acc[NT] = {};

    const int mA   = lane & 15;   // row (A) / col (B) within tile
    const int ksel = lane >> 4;   // which K-pair this half-wave holds

    // per-nt B base pointers: Wt row (nt*16+mA), K-offset ksel*2
    const float* pB[NT];
#pragma unroll
    for (int nt = 0; nt < NT; ++nt)
        pB[nt] = Wt + (size_t)(nt * 16 + mA) * K + ksel * 2;

    // cooperative A staging: lane pair covers one row, 32 contiguous floats
    const int rL   = lane >> 1;
    const int half = lane & 1;
    float* sMine = sA[wave];

    for (int kc = 0; kc < K; kc += 64) {
        const float4* aRow = (const float4*)(A + (size_t)(m0 + rL) * K + kc + half * 32);
        float4* sRow = (float4*)(sMine + rL * 64 + half * 32);
        float4 t0 = aRow[0], t1 = aRow[1], t2 = aRow[2], t3 = aRow[3];
        float4 t4 = aRow[4], t5 = aRow[5], t6 = aRow[6], t7 = aRow[7];
        sRow[0] = t0; sRow[1] = t1; sRow[2] = t2; sRow[3] = t3;
        sRow[4] = t4; sRow[5] = t5; sRow[6] = t6; sRow[7] = t7;
        // per-wave LDS region; DS ops are in-order within a wave -> no barrier

        const float* sAf = sMine + mA * 64 + ksel * 2;
#pragma unroll 4
        for (int kk = 0; kk < 16; ++kk) {
            v2f a = *(const v2f*)(sAf + kk * 4);          // ds_load_b64
#pragma unroll
            for (int nt = 0; nt < NT; ++nt) {
                v2f b = *(const v2f*)(pB[nt] + kc + kk * 4);  // global_load_b64
                acc[nt] = __builtin_amdgcn_wmma_f32_16x16x4_f32(
                    false, a, false, b, (short)0, acc[nt], false, false);
            }
        }
    }

    // epilogue: fuse per-row norm scale
    const int rowLo = m0 + ksel * 8;
#pragma unroll
    for (int r = 0; r < 8; ++r) {
        const int row = rowLo + r;
        if (row < M) {
            const float nrm = normv[row];
#pragma unroll
            for (int nt = 0; nt < NT; ++nt)
                out[(size_t)row * Npad + nt * 16 + mA] = acc[nt][r] * nrm;
        }
    }
}

// ---------------------------------------------------------------------------
// Edge scatter: one wave per edge. Coalesced float4 gather of nh[src] row
// (L2-resident: <=51.2 MB < 192 MB L2), f32 atomic adds into agg[dst].
// ---------------------------------------------------------------------------
__global__ __launch_bounds__(256) void gcn_edge_scatter(
    const float* __restrict__ nh, const int* __restrict__ src,
    const int* __restrict__ dst, float* __restrict__ agg, int E, int F)
{
    const int lane = threadIdx.x & 31;
    const int e = (int)((blockIdx.x * blockDim.x + threadIdx.x) >> 5);
    if (e >= E) return;
    const int s = __builtin_amdgcn_readfirstlane(src[e]);
    const int d = __builtin_amdgcn_readfirstlane(dst[e]);
    const int f = lane * 4;
    if (f < F) {
        float4 v = *(const float4*)(nh + (size_t)s * F + f);
        float* po = agg + (size_t)d * F + f;
        atomicAdd(po + 0, v.x);
        atomicAdd(po + 1, v.y);
        atomicAdd(po + 2, v.z);
        atomicAdd(po + 3, v.w);
    }
}

// ---------------------------------------------------------------------------
// Zero-fill (graph-capture-safe replacement for hipMemsetAsync)
// ---------------------------------------------------------------------------
__global__ void gcn_zero_f4(float4* __restrict__ p, long n4)
{
    long i = (long)blockIdx.x * blockDim.x + threadIdx.x;
    if (i < n4) p[i] = make_float4(0.f, 0.f, 0.f, 0.f);
}

// ---------------------------------------------------------------------------
// h = relu(h + b), in place (F = 128, power of two -> cheap modulo)
// ---------------------------------------------------------------------------
__global__ void gcn_bias_relu(float* __restrict__ h, const float* __restrict__ b,
                              long total, int F)
{
    long i = (long)blockIdx.x * blockDim.x + threadIdx.x;
    if (i >= total) return;
    float v = h[i] + b[(int)(i % F)];
    h[i] = v > 0.f ? v : 0.f;
}

// ---------------------------------------------------------------------------
// Final: out[m, c] = agg[m, c] + b[c]   (strip padding, no relu)
// ---------------------------------------------------------------------------
__global__ void gcn_bias_out(const float* __restrict__ agg,
                             const float* __restrict__ b,
                             float* __restrict__ out, long total, int C, int Fpad)
{
    long i = (long)blockIdx.x * blockDim.x + threadIdx.x;
    if (i >= total) return;
    long m = i / C;
    int c = (int)(i - m * C);
    out[i] = agg[m * (long)Fpad + c] + b[c];
}

extern "C" void kernel_launch(void* const* d_in, const int* in_sizes, int n_in,
                              void* d_out, int out_size, void* d_ws, size_t ws_size,
                              hipStream_t stream)
{
    (void)n_in; (void)out_size; (void)ws_size;

    const float* features = (const float*)d_in[0];
    const float* normv    = (const float*)d_in[1];
    const float* W0       = (const float*)d_in[2];
    const float* b0       = (const float*)d_in[3];
    const float* W1       = (const float*)d_in[4];
    const float* b1       = (const float*)d_in[5];
    const float* W2       = (const float*)d_in[6];
    const float* b2       = (const float*)d_in[7];
    const int*   src      = (const int*)d_in[8];
    const int*   dst      = (const int*)d_in[9];
    float*       outp     = (float*)d_out;

    const int Nn = in_sizes[1];             // 100000 nodes
    const int E  = in_sizes[8];             // 1.6M edges
    const int K0 = in_sizes[0] / Nn;        // 256
    const int H  = in_sizes[3];             // 128
    const int C  = in_sizes[7];             // 47
    const int Cpad = 48;                    // NT=3

    float* buf0 = (float*)d_ws;             // nh buffer   [Nn, 128]
    float* buf1 = buf0 + (size_t)Nn * H;    // agg buffer  [Nn, 128]
    float* Wt0  = buf1 + (size_t)Nn * H;    // [128, 256]
    float* Wt1  = Wt0 + (size_t)H * K0;     // [128, 128]
    float* Wt2  = Wt1 + (size_t)H * H;      // [48, 128]

    const int mTiles = (Nn + 15) >> 4;
    const int gemmBlocks = (mTiles + 3) / 4;
    const int edgeBlocks = (int)(((long)E * 32 + 255) / 256);
    auto eblk = [](long n) { return (int)((n + 255) / 256); };

    const long nhH = (long)Nn * H;          // elements, width 128
    const long nhC = (long)Nn * Cpad;       // elements, width 48

    // weight transposes (tiny; W stays resident in L0/L2 during GEMMs)
    gcn_transpose_pad<<<eblk((long)H * K0), 256, 0, stream>>>(W0, Wt0, K0, H, H);
    gcn_transpose_pad<<<eblk((long)H * H),  256, 0, stream>>>(W1, Wt1, H, H, H);
    gcn_transpose_pad<<<eblk((long)Cpad * H), 256, 0, stream>>>(W2, Wt2, H, C, Cpad);

    // ---- Layer 0: 256 -> 128, relu -------------------------------------
    gcn_gemm_norm_wmma<8><<<gemmBlocks, 128, 0, stream>>>(features, Wt0, normv, buf0, Nn, K0);
    gcn_zero_f4<<<eblk(nhH / 4), 256, 0, stream>>>((float4*)buf1, nhH / 4);
    gcn_edge_scatter<<<edgeBlocks, 256, 0, stream>>>(buf0, src, dst, buf1, E, H);
    gcn_bias_relu<<<eblk(nhH), 256, 0, stream>>>(buf1, b0, nhH, H);

    // ---- Layer 1: 128 -> 128, relu -------------------------------------
    gcn_gemm_norm_wmma<8><<<gemmBlocks, 128, 0, stream>>>(buf1, Wt1, normv, buf0, Nn, H);
    gcn_zero_f4<<<eblk(nhH / 4), 256, 0, stream>>>((float4*)buf1, nhH / 4);
    gcn_edge_scatter<<<edgeBlocks, 256, 0, stream>>>(buf0, src, dst, buf1, E, H);
    gcn_bias_relu<<<eblk(nhH), 256, 0, stream>>>(buf1, b1, nhH, H);

    // ---- Layer 2: 128 -> 47 (padded to 48), no relu ---------------------
    gcn_gemm_norm_wmma<3><<<gemmBlocks, 128, 0, stream>>>(buf1, Wt2, normv, buf0, Nn, H);
    gcn_zero_f4<<<eblk(nhC / 4), 256, 0, stream>>>((float4*)buf1, nhC / 4);
    gcn_edge_scatter<<<edgeBlocks, 256, 0, stream>>>(buf0, src, dst, buf1, E, Cpad);
    gcn_bias_out<<<eblk((long)Nn * C), 256, 0, stream>>>(buf1, b2, outp, (long)Nn * C, C, Cpad);
}